// MemoryCore_31353261261521
// MI455X (gfx1250) — compile-verified
//
#include <hip/hip_runtime.h>
#include <stdint.h>

typedef float v2f __attribute__((ext_vector_type(2)));
typedef float v8f __attribute__((ext_vector_type(8)));

// D = A(16x4 f32) * B(4x16 f32) + C(16x16 f32), full fp32 precision.
#define WMMA_F32_16x16x4(a, b, c) \
  __builtin_amdgcn_wmma_f32_16x16x4_f32(false, (a), false, (b), 0, (c), false, false)

// Shapes fixed by the reference.
#define BQ   8
#define TT   4
#define DE   128
#define DO   512
#define HW   1024
#define THW  4096
#define SCALE 0.08838834764831845f   // 1/sqrt(128)

// LDS tile row pitch in floats: 288B rows -> 16B aligned for b128 async
// writes; 72 % 64 == 8 bank shift per row, so the (row, row+2) pair read by
// the two half-waves of one ds_load hits disjoint bank sets.
#define PITCH 72
// K-chunk for GEMM2 LDS pipelining.
#define KC 64

// Issue one async global->LDS 128-bit copy. VDST = LDS byte address,
// VADDR = 32-bit byte offset, SADDR = 64-bit base. Tracked on ASYNCcnt.
__device__ __forceinline__ void async_cp_b128(uint32_t lds_addr, uint32_t voff,
                                              const float* sbase) {
  asm volatile("global_load_async_to_lds_b128 %0, %1, %2"
               :: "v"(lds_addr), "v"(voff), "s"(sbase) : "memory");
}

// ---------------------------------------------------------------------------
// Kernel 1: p_raw[b,k,n] = (1/sqrt(De)) * sum_d mi[b,k,d] * qi[b,d,n]
//   mi[b,k,d] = m_in[b, k>>10, d, k&1023]   (16-row k-tile stays in one t)
// Block = 4 waves sharing (b, n-strip); the 128x64 q_in panel is staged once
// into LDS via async b128 copies and reused by all 4 k-tiles.
// ---------------------------------------------------------------------------
__global__ __launch_bounds__(128) void scores_wmma(
    const float* __restrict__ m_in, const float* __restrict__ q_in,
    float* __restrict__ p_out) {
  __shared__ float sh[DE * PITCH];            // 36864 B

  const int tid  = threadIdx.x;
  const int lane = tid & 31;
  const int wave = tid >> 5;

  // blocks: 8 b * 64 kgroups * 16 nstrips = 8192
  const int nstrip = blockIdx.x & 15;
  const int kgrp   = (blockIdx.x >> 4) & 63;
  const int b      = blockIdx.x >> 10;

  const int ktile = kgrp * 4 + wave;
  const int k0  = ktile << 4;
  const int t   = k0 >> 10;
  const int hw0 = k0 & 1023;
  const int n0  = nstrip << 6;

  const int mrow = lane & 15;                 // A row / B col / D col
  const int kb   = (lane >> 4) << 1;          // K pair base: 0 or 2

  // ---- stage B panel: q_in[b, 0..127, n0..n0+63] -> LDS (async b128) ----
  const uint32_t shbase = (uint32_t)(uintptr_t)(void*)&sh[0];
  const float* G = q_in + ((size_t)b * DE) * HW + n0;
  const int nq  = tid & 15;                   // float4 column 0..15
  const int kr0 = tid >> 4;                   // row group 0..7
#pragma unroll
  for (int i = 0; i < 16; ++i) {
    const int dr = kr0 + i * 8;               // 0..127
    async_cp_b128(shbase + (uint32_t)(dr * PITCH + nq * 4) * 4,
                  (uint32_t)(dr * HW + nq * 4) * 4, G);
  }
  asm volatile("s_wait_asynccnt 0x0" ::: "memory");
  __syncthreads();

  // ---- compute ----
  // A[m, d] = m_in[((b*4 + t)*128 + d)*1024 + hw0 + m]
  const float* Abase = m_in + (((size_t)(b * 4 + t) * DE) << 10) + hw0 + mrow;

  v8f acc0 = {}, acc1 = {}, acc2 = {}, acc3 = {};
#pragma unroll 4
  for (int d0 = 0; d0 < DE; d0 += 4) {
    v2f a;
    a.x = Abase[(size_t)(d0 + kb) << 10];
    a.y = Abase[(size_t)(d0 + kb + 1) << 10];
    const int r0 = (d0 + kb) * PITCH + mrow;
    const int r1 = r0 + PITCH;
    v2f b0 = {sh[r0 +  0], sh[r1 +  0]};
    v2f b1 = {sh[r0 + 16], sh[r1 + 16]};
    v2f b2 = {sh[r0 + 32], sh[r1 + 32]};
    v2f b3 = {sh[r0 + 48], sh[r1 + 48]};
    acc0 = WMMA_F32_16x16x4(a, b0, acc0);
    acc1 = WMMA_F32_16x16x4(a, b1, acc1);
    acc2 = WMMA_F32_16x16x4(a, b2, acc2);
    acc3 = WMMA_F32_16x16x4(a, b3, acc3);
  }

  // D vgpr i -> row k0 + i + 8*(lane>=16), col n0 + (lane&15)
  float* out = p_out + ((size_t)b * THW + k0 + ((lane >> 4) << 3)) * HW +
               n0 + mrow;
#pragma unroll
  for (int i = 0; i < 8; ++i) {
    out[(size_t)i * HW +  0] = acc0[i] * SCALE;
    out[(size_t)i * HW + 16] = acc1[i] * SCALE;
    out[(size_t)i * HW + 32] = acc2[i] * SCALE;
    out[(size_t)i * HW + 48] = acc3[i] * SCALE;
  }
}

// ---------------------------------------------------------------------------
// Kernel 2: in-place numerically-stable softmax along k (stride HW floats).
// ---------------------------------------------------------------------------
__global__ __launch_bounds__(256) void softmax_k(float* __restrict__ p) {
  const int c = blockIdx.x * 256 + threadIdx.x;  // 0..8191
  const int b = c >> 10;
  const int n = c & 1023;
  float* col = p + ((size_t)b * THW) * HW + n;

  float m = -INFINITY;
  for (int k = 0; k < THW; ++k) m = fmaxf(m, col[(size_t)k << 10]);
  float s = 0.f;
  for (int k = 0; k < THW; ++k) {
    float e = __expf(col[(size_t)k << 10] - m);
    col[(size_t)k << 10] = e;
    s += e;
  }
  const float inv = 1.0f / s;
  for (int k = 0; k < THW; ++k) col[(size_t)k << 10] *= inv;
}

// ---------------------------------------------------------------------------
// Kernel 3: mem[b,d,n] = sum_k mo[b,d,k] * p[b,k,n]
// Block = 4 waves sharing (b, n-strip), covering 64 d rows. The shared
// 64K x 64n p-tile is double-buffered in LDS with one-chunk-ahead async
// prefetch (ASYNCcnt pipelining); A rows are contiguous 8B global loads.
// ---------------------------------------------------------------------------
__global__ __launch_bounds__(128) void applymem_wmma(
    const float* __restrict__ m_out, const float* __restrict__ p,
    float* __restrict__ mem) {
  __shared__ float sh[2 * KC * PITCH];        // 2 x 18432 B

  const int tid  = threadIdx.x;
  const int lane = tid & 31;
  const int wave = tid >> 5;

  // blocks: 8 b * 8 dgroups * 16 nstrips = 1024
  const int nstrip = blockIdx.x & 15;
  const int dgrp   = (blockIdx.x >> 4) & 7;
  const int b      = blockIdx.x >> 7;

  const int d0 = (dgrp * 4 + wave) << 4;      // this wave's 16 d rows
  const int n0 = nstrip << 6;

  const int mrow = lane & 15;
  const int kb   = (lane >> 4) << 1;

  const float* Arow  = m_out + (size_t)b * DO * THW +
                       (size_t)(d0 + mrow) * THW + kb;
  const float* Gbase = p + ((size_t)b * THW) * HW + n0;

  const uint32_t shbase = (uint32_t)(uintptr_t)(void*)&sh[0];
  const int nq  = tid & 15;
  const int kr0 = tid >> 4;

  // Stage one 64K x 64n chunk of p into LDS buffer `buf` (8 b128 / thread).
  auto stage = [&](int kc, int buf) {
    const float* g = Gbase + (size_t)kc * HW;
    const uint32_t lbase = shbase + (uint32_t)buf * (KC * PITCH * 4);
#pragma unroll
    for (int i = 0; i < 8; ++i) {
      const int kr = kr0 + i * 8;             // 0..63
      async_cp_b128(lbase + (uint32_t)(kr * PITCH + nq * 4) * 4,
                    (uint32_t)(kr * HW + nq * 4) * 4, g);
    }
  };

  v8f acc0 = {}, acc1 = {}, acc2 = {}, acc3 = {};

  stage(0, 0);
  for (int c = 0; c < THW / KC; ++c) {        // 64 chunks
    if (c + 1 < THW / KC) {
      stage((c + 1) * KC, (c + 1) & 1);
      // 8 newer copies in flight; <=8 outstanding => chunk c fully in LDS
      asm volatile("s_wait_asynccnt 0x8" ::: "memory");
    } else {
      asm volatile("s_wait_asynccnt 0x0" ::: "memory");
    }
    __syncthreads();                          // all waves' chunk-c data ready

    const float* shb = sh + (c & 1) * (KC * PITCH);
    const float* Ac  = Arow + c * KC;
#pragma unroll 4
    for (int s = 0; s < KC; s += 4) {
      v2f a = *reinterpret_cast<const v2f*>(Ac + s);   // 8B-aligned pair
      const int r0 = (s + kb) * PITCH + mrow;
      const int r1 = r0 + PITCH;
      v2f b0 = {shb[r0 +  0], shb[r1 +  0]};
      v2f b1 = {shb[r0 + 16], shb[r1 + 16]};
      v2f b2 = {shb[r0 + 32], shb[r1 + 32]};
      v2f b3 = {shb[r0 + 48], shb[r1 + 48]};
      acc0 = WMMA_F32_16x16x4(a, b0, acc0);
      acc1 = WMMA_F32_16x16x4(a, b1, acc1);
      acc2 = WMMA_F32_16x16x4(a, b2, acc2);
      acc3 = WMMA_F32_16x16x4(a, b3, acc3);
    }
    __syncthreads();                          // done reading buf (c&1)
  }

  // mem_out layout: (B, 2*Do, HW); mem occupies rows [0, 512)
  float* out = mem + ((size_t)b * (2 * DO) + d0 + ((lane >> 4) << 3)) * HW +
               n0 + mrow;
#pragma unroll
  for (int i = 0; i < 8; ++i) {
    out[(size_t)i * HW +  0] = acc0[i];
    out[(size_t)i * HW + 16] = acc1[i];
    out[(size_t)i * HW + 32] = acc2[i];
    out[(size_t)i * HW + 48] = acc3[i];
  }
}

// ---------------------------------------------------------------------------
// Kernel 4: mem_out[b, 512+d, hw] = q_out[b, d, hw]  (float4 copy)
// ---------------------------------------------------------------------------
__global__ __launch_bounds__(256) void concat_qout(
    const float4* __restrict__ q_out, float4* __restrict__ out) {
  const size_t i = (size_t)blockIdx.x * 256 + threadIdx.x;  // float4 units
  const size_t per_b = (size_t)DO * HW / 4;                 // 131072
  const size_t b = i / per_b;
  const size_t r = i % per_b;
  out[b * ((size_t)2 * DO * HW / 4) + per_b + r] = q_out[i];
}

extern "C" void kernel_launch(void* const* d_in, const int* in_sizes, int n_in,
                              void* d_out, int out_size, void* d_ws, size_t ws_size,
                              hipStream_t stream) {
  const float* m_in  = (const float*)d_in[0];
  const float* m_out = (const float*)d_in[1];
  const float* q_in  = (const float*)d_in[2];
  const float* q_out = (const float*)d_in[3];

  float* out = (float*)d_out;
  float* mem = out;                                       // (8, 1024, 1024)
  float* p   = out + (size_t)BQ * (2 * DO) * HW;          // (8, 4096, 1024)

  // GEMM1: 8 b * 64 kgroups * 16 nstrips = 8192 blocks, 4 waves each
  scores_wmma<<<8192, 128, 0, stream>>>(m_in, q_in, p);
  // Softmax: 8192 columns
  softmax_k<<<32, 256, 0, stream>>>(p);
  // GEMM2: 8 b * 8 dgroups * 16 nstrips = 1024 blocks, 4 waves each
  applymem_wmma<<<1024, 128, 0, stream>>>(m_out, p, mem);
  // Concat: 8*512*1024/4 = 1048576 float4
  concat_qout<<<4096, 256, 0, stream>>>((const float4*)q_out, (float4*)out);
}